// NeedlemanWunschDecoder_27161373180429
// MI455X (gfx1250) — compile-verified
//
#include <hip/hip_runtime.h>

// ---------------------------------------------------------------------------
// Smoothed Needleman-Wunsch (softmax-max DP), 2048x2048, scalar f32 output.
//
// Latency-bound problem (16MB theta => 0.7us of HBM time at 23.3 TB/s; the
// ~4095-step serial dependency chain dominates). Strategy:
//  * skewed wavefront: one wave32 wave per 32-row strip, lane = row,
//    recurrence kept entirely in VGPRs
//  * neighbor values moved with ds_bpermute lane shifts (wave32)
//  * strip->strip handoff through LDS ring buffers, s_sleep polling,
//    s_wait_dscnt release (gfx1250 split counters)
//  * theta double-buffered in registers one 32-col chunk ahead
//  * gfx1250 async engine: GLOBAL_LOAD_ASYNC_TO_LDS_B128 prefetches the
//    theta stream ~3 chunks ahead into an LDS scratch (ASYNCcnt-tracked,
//    throttled with s_wait_asynccnt), plus global_prefetch_b8 at +2 chunks.
// ---------------------------------------------------------------------------

#define NEGV  (-1.0e10f)
#define NN    2048          // N == M == 2048
#define RING  256           // ring columns per wave (power of two)
#define RMASK (RING - 1)
#define G     32            // chunk length (iterations between syncs)
#define NCHUNK 65           // t = 1 .. 2080 (last iteration is a masked pad)

// lane shift by 1 (lane l receives lane l-1's value); lane 0 garbage (fixed up)
__device__ __forceinline__ float shfl_up1(float x) {
  int lane = (int)(threadIdx.x & 31);
  int r = __builtin_amdgcn_ds_bpermute(((lane - 1) & 31) << 2, __float_as_int(x));
  return __int_as_float(r);
}

// broadcast lane k's value to all lanes (k is a compile-time constant here)
__device__ __forceinline__ float bcast(float x, int k) {
  return __int_as_float(__builtin_amdgcn_readlane(__float_as_int(x), k));
}

// gfx1250 async engine: copy 16B from global into a per-lane LDS scratch slot.
// Pure prefetch role (fills WGP$/L2 on the way); tracked by ASYNCcnt.
__device__ __forceinline__ void async_prefetch_b128(unsigned ldsOff,
                                                    const float* gptr) {
  asm volatile("global_load_async_to_lds_b128 %0, %1, off"
               :: "v"(ldsOff), "v"((unsigned long long)gptr)
               : "memory");
}

// MIN: 0 = synthetic top boundary (strip 0), 1 = LDS ring from prev wave,
//      2 = full bottom-row buffer (phase-1 wave 0)
// MOUT: 0 = LDS ring to next wave, 1 = full bottom-row buffer (phase-0 wave 31),
//       2 = none / capture final output (phase-1 wave 31)
template <int MIN, int MOUT>
__device__ __forceinline__ void run_strip(
    int strip, int lane,
    const float* __restrict__ theta, float A,
    float* ringIn, float* ringOut, float* bottom,
    volatile int* progPrev, volatile int* progNext, volatile int* progMine,
    unsigned myScratchOff,
    float* __restrict__ outp)
{
  const float* rowPtr = theta + (size_t)(strip * 32 + lane) * NN;

  float vPrev  = NEGV;   // V[i, j-1]  (this lane, iteration t-1)
  float vPrev2 = NEGV;   // V[i, j-2]  (this lane, iteration t-2)
  float btm1   = (MIN == 0) ? 0.0f : NEGV;  // B[t-1] carry; B[0]=0 only strip 0

  // register double buffer for theta (32 values / lane / chunk)
  float thNext[G];
  {
    int base = -lane;  // chunk 0 columns: t-l-1 for t = 1..32
#pragma unroll
    for (int k = 0; k < G; ++k) {
      int c = base + k;
      c = c < 0 ? 0 : (c > NN - 1 ? NN - 1 : c);
      thNext[k] = rowPtr[c];
    }
  }

  for (int ch = 0; ch < NCHUNK; ++ch) {
    const int t0 = 1 + ch * G;

    // -- consumer gate: producer must have written boundary cols <= t0+G-1
    if constexpr (MIN == 1) {
      int hi = t0 + G - 1;
      if (hi > NN) hi = NN;
      const int need = hi + 31;  // producer iteration that wrote column `hi`
      while (*progPrev < need) __builtin_amdgcn_s_sleep(1);
    }
    // -- producer back-pressure: don't overwrite ring cols consumer still needs
    if constexpr (MOUT == 0) {
      const int need = t0 - RING + 1;  // max col written this chunk is t0
      if (need > 0) {
        while (*progNext < need) __builtin_amdgcn_s_sleep(1);
      }
    }
    asm volatile("" ::: "memory");

    // -- boundary chunk: lane l holds B[t0+l]
    float bb;
    if constexpr (MIN == 0)      bb = NEGV;
    else if constexpr (MIN == 1) bb = ringIn[(t0 + lane) & RMASK];
    else                         bb = bottom[t0 + lane];

    // -- theta: consume prefetched chunk, issue next chunk, warm ahead
    float thCur[G];
#pragma unroll
    for (int k = 0; k < G; ++k) thCur[k] = thNext[k];
    {
      int base = t0 + G - 1 - lane;
#pragma unroll
      for (int k = 0; k < G; ++k) {
        int c = base + k;
        c = c < 0 ? 0 : (c > NN - 1 ? NN - 1 : c);
        thNext[k] = rowPtr[c];
      }
      // cacheline prefetch 2 chunks ahead (gfx1250 global_prefetch_b8)
      int pc = base + 2 * G;
      pc = pc < 0 ? 0 : (pc > NN - 1 ? NN - 1 : pc);
      __builtin_prefetch(rowPtr + pc, 0, 1);
      // async-engine prefetch 3 chunks ahead (16B, 16B-aligned column)
      int pa = (base + 3 * G) & ~3;
      pa = pa < 0 ? 0 : (pa > NN - 4 ? NN - 4 : pa);
      async_prefetch_b128(myScratchOff, rowPtr + pa);
      // throttle ASYNCcnt (1 issue per chunk -> counter stays small)
      asm volatile("s_wait_asynccnt 0x8" ::: "memory");
    }

    // -- 32 dependent wavefront steps
#pragma unroll
    for (int k = 0; k < G; ++k) {
      const int t = t0 + k;
      float bt = (MIN == 0) ? NEGV : bcast(bb, k);   // B[t] (uniform)
      float up = shfl_up1(vPrev);    // V[i-1, j]   from lane-1 @ t-1
      float dg = shfl_up1(vPrev2);   // V[i-1, j-1] from lane-1 @ t-2
      if (lane == 0) { up = bt; dg = btm1; }
      btm1 = bt;

      const int j = t - lane;
      const float xu = up + A;
      const float xl = vPrev + A;    // left = V[i, j-1]
      float m = fmaxf(fmaxf(xu, dg), xl);
      float s = __expf(xu - m) + __expf(dg - m) + __expf(xl - m);
      float v = thCur[k] + m + __logf(s);
      v = (j >= 1 && j <= NN) ? v : NEGV;

      vPrev2 = vPrev;
      vPrev  = v;

      // -- handoff: lane 31's value is the strip's bottom row at column t-31
      const int jj = t - 31;
      if constexpr (MOUT == 0) {
        if (lane == 31 && jj >= 1 && jj <= NN) ringOut[jj & RMASK] = v;
      } else if constexpr (MOUT == 1) {
        if (lane == 31 && jj >= 1 && jj <= NN) bottom[jj] = v;
      } else {
        if (lane == 31 && jj == NN) *outp = v;  // V[2048, 2048]
      }
    }

    // -- publish progress (DS ops are wave-in-order; explicit split-counter
    //    wait makes the release ordering explicit on gfx1250)
    asm volatile("s_wait_dscnt 0" ::: "memory");
    if (lane == 0) *progMine = t0 + G - 1;
  }
}

__global__ void __launch_bounds__(1024, 1)
nw_wavefront_kernel(const float* __restrict__ theta,
                    const float* __restrict__ Ap,
                    float* __restrict__ out)
{
  __shared__ float ring[32][RING];   // 32 KB: per-wave bottom-row rings
  __shared__ float bottom[2112];     // 8.25 KB: full row V[1024, *] across phases
  __shared__ int   prog[32];         // per-wave progress (iterations completed)
  __shared__ __align__(16) char asyncScratch[1024 * 16];  // 16 KB async landing pad

  const int tid  = (int)threadIdx.x;
  const int lane = tid & 31;
  const int w    = tid >> 5;
  const float A  = Ap[0];

  // low 32 bits of the shared-aperture generic address == LDS byte offset
  const unsigned myScratchOff =
      (unsigned)(size_t)(&asyncScratch[(size_t)tid * 16]);

  if (tid < 32) prog[tid] = 0;
  if (tid == 0) bottom[0] = NEGV;    // V[1024, 0]
  __syncthreads();

  // ---- phase 0: strips 0..31 (rows 1..1024 of V) ----
  if (w == 0)
    run_strip<0, 0>(0, lane, theta, A, nullptr, ring[0], bottom,
                    nullptr, &prog[1], &prog[0], myScratchOff, out);
  else if (w == 31)
    run_strip<1, 1>(31, lane, theta, A, ring[30], nullptr, bottom,
                    &prog[30], nullptr, &prog[31], myScratchOff, out);
  else
    run_strip<1, 0>(w, lane, theta, A, ring[w - 1], ring[w], bottom,
                    &prog[w - 1], &prog[w + 1], &prog[w], myScratchOff, out);

  __syncthreads();
  if (tid < 32) prog[tid] = 0;
  __syncthreads();

  // ---- phase 1: strips 32..63 (rows 1025..2048 of V) ----
  if (w == 0)
    run_strip<2, 0>(32, lane, theta, A, nullptr, ring[0], bottom,
                    nullptr, &prog[1], &prog[0], myScratchOff, out);
  else if (w == 31)
    run_strip<1, 2>(63, lane, theta, A, ring[30], nullptr, bottom,
                    &prog[30], nullptr, &prog[31], myScratchOff, out);
  else
    run_strip<1, 0>(32 + w, lane, theta, A, ring[w - 1], ring[w], bottom,
                    &prog[w - 1], &prog[w + 1], &prog[w], myScratchOff, out);

  // drain the async prefetch engine before wave exit
  asm volatile("s_wait_asynccnt 0x0" ::: "memory");
}

extern "C" void kernel_launch(void* const* d_in, const int* in_sizes, int n_in,
                              void* d_out, int out_size, void* d_ws, size_t ws_size,
                              hipStream_t stream) {
  (void)in_sizes; (void)n_in; (void)out_size; (void)d_ws; (void)ws_size;
  const float* theta = (const float*)d_in[0];
  const float* A     = (const float*)d_in[1];  // scalar gap penalty
  nw_wavefront_kernel<<<1, 1024, 0, stream>>>(theta, A, (float*)d_out);
}